// ChamferLoss_377957122587
// MI455X (gfx1250) — compile-verified
//
#include <hip/hip_runtime.h>

typedef __attribute__((ext_vector_type(2))) float v2f;
typedef __attribute__((ext_vector_type(8))) float v8f;

#define NPTS  4096
#define NB    8
#define TILES (NPTS / 16)

// One wave owns one 16-row tile of X for one batch, sweeps all 256 16-col
// tiles of Y. Each tile is one v_wmma_f32_16x16x4_f32:
//   A row i = [-2x0, -2x1, -2x2, 1]     (loop-invariant)
//   B col j = [y0, y1, y2, |y_j|^2]
//   C       = 0                          (loop-invariant)
//   => D = |y_j|^2 - 2 x.y
// |x_i|^2 is constant along j so it is added once AFTER the min over j:
//   min_j pdist = |x_i|^2 + min_j (|y_j|^2 - 2 x.y)
// Running element-wise min lives in 8 VGPRs; one shfl_xor min-tree at the
// end collapses the column dimension.
__global__ __launch_bounds__(256) void chamfer_rowmin_kernel(
    const float* __restrict__ X, const float* __restrict__ Y,
    float* __restrict__ rowmin)
{
    const int lane  = threadIdx.x & 31;
    const int wave  = threadIdx.x >> 5;
    const int itile = blockIdx.x * 8 + wave;   // 32 blocks * 8 waves = 256 tiles
    const int b     = blockIdx.y;
    const int m     = lane & 15;               // row/col index within tile
    const bool hi   = (lane >= 16);            // second 16-lane group -> K=2,3

    // ---- A matrix (16x4 f32, 2 VGPRs): lanes<16 K=0/1, lanes>=16 K=2/3 ----
    const float* xp = X + ((size_t)b * NPTS + (size_t)itile * 16 + m) * 3;
    float x0 = xp[0], x1 = xp[1], x2c = xp[2];
    float xsq = x0 * x0 + x1 * x1 + x2c * x2c;
    v2f A;
    A.x = hi ? (-2.0f * x2c) : (-2.0f * x0);   // K=2 : K=0
    A.y = hi ? 1.0f          : (-2.0f * x1);   // K=3 : K=1

    const v8f Czero = {};                      // loop-invariant C = 0

    v8f acc;
#pragma unroll
    for (int r = 0; r < 8; ++r) acc[r] = 3.0e38f;

    const float* yb = Y + (size_t)b * NPTS * 3;
    for (int jt = 0; jt < TILES; ++jt) {
        const float* yp = yb + (size_t)(jt * 16 + m) * 3;
        float y0 = yp[0], y1 = yp[1], y2c = yp[2];
        float ysq = y0 * y0 + y1 * y1 + y2c * y2c;

        // B matrix (4x16 f32, 2 VGPRs): lanes<16 rows K=0/1, lanes>=16 K=2/3
        v2f B;
        B.x = hi ? y2c : y0;
        B.y = hi ? ysq : y1;

        v8f D = __builtin_amdgcn_wmma_f32_16x16x4_f32(
            /*neg_a=*/false, A, /*neg_b=*/false, B,
            /*c_mod=*/(short)0, Czero, /*reuse_a=*/false, /*reuse_b=*/false);

#pragma unroll
        for (int r = 0; r < 8; ++r) acc[r] = fminf(acc[r], D[r]);
    }

    // Min over columns n (xor masks < 16 stay within each 16-lane group)
#pragma unroll
    for (int mask = 1; mask <= 8; mask <<= 1) {
#pragma unroll
        for (int r = 0; r < 8; ++r) {
            float o = __shfl_xor(acc[r], mask, 32);
            acc[r]  = fminf(acc[r], o);
        }
    }

    // Add |x_i|^2 for the row each acc[r] represents (row = r + 8*hi).
    // All lanes run the shuffles so ds_bpermute sources stay EXEC-active.
#pragma unroll
    for (int r = 0; r < 8; ++r) {
        float xq = __shfl(xsq, (hi ? 8 : 0) + r, 32);
        acc[r] += xq;
    }

    // D-layout: VGPR r holds row r (lanes<16) / row r+8 (lanes>=16)
    if ((lane & 15) == 0) {
        float* out = rowmin + (size_t)b * NPTS + (size_t)itile * 16 + (hi ? 8 : 0);
#pragma unroll
        for (int r = 0; r < 8; ++r) out[r] = acc[r];
    }
}

// Final reduction: sum of both min arrays (2 * 8 * 4096 floats); both means
// share the same 1/32768 denominator, then * 0.005.
__global__ __launch_bounds__(1024) void chamfer_reduce_kernel(
    const float* __restrict__ mins, float* __restrict__ out)
{
    __shared__ float s[1024];
    float sum = 0.0f;
    const int total = 2 * NB * NPTS;
    for (int k = threadIdx.x; k < total; k += 1024) sum += mins[k];
    s[threadIdx.x] = sum;
    __syncthreads();
    for (int w = 512; w > 0; w >>= 1) {
        if ((int)threadIdx.x < w) s[threadIdx.x] += s[threadIdx.x + w];
        __syncthreads();
    }
    if (threadIdx.x == 0)
        out[0] = s[0] * (0.005f / (float)(NB * NPTS));
}

extern "C" void kernel_launch(void* const* d_in, const int* in_sizes, int n_in,
                              void* d_out, int out_size, void* d_ws, size_t ws_size,
                              hipStream_t stream) {
    const float* x = (const float*)d_in[0];
    const float* y = (const float*)d_in[1];
    float* mins  = (float*)d_ws;             // [0 .. 32768)  min over j per (b,i)
    float* minsT = mins + NB * NPTS;         // [32768..65536) min over i per (b,j)

    dim3 grid(TILES / 8, NB);
    // min over axis=2 (over j): rowmin of pdist(x, y)
    chamfer_rowmin_kernel<<<grid, 256, 0, stream>>>(x, y, mins);
    // min over axis=1 (over i): rowmin of pdist(y, x)  (pdist transposes under swap)
    chamfer_rowmin_kernel<<<grid, 256, 0, stream>>>(y, x, minsT);

    chamfer_reduce_kernel<<<1, 1024, 0, stream>>>(mins, (float*)d_out);
}